// DetNet_62414464745531
// MI455X (gfx1250) — compile-verified
//
#include <hip/hip_runtime.h>

// ---- problem constants (match reference) ----
constexpr int D_LAYERS = 10;
constexpr int D_BL  = 256;   // BLOCK_LEN
constexpr int D_SL  = 264;   // SYM_LEN
constexpr int D_VL  = 128;   // V_LEN
constexpr int D_ZL  = 512;   // Z_LEN
constexpr int D_IN1 = D_SL + D_VL;      // 392
constexpr int D_2SL = 2 * D_SL;         // 528
constexpr int D_PR  = 2 * D_BL;         // 512 Psi rows
constexpr int NTHR  = 256;              // 8 wave32 per block

typedef __attribute__((ext_vector_type(2))) float v2f;
typedef __attribute__((ext_vector_type(8))) float v8f;

struct Args {
  const float *y_e, *y_o, *Psi_e, *Psi_o, *mapp_mag, *mapp_phase;
  const float *W1m, *b1m, *W2m, *b2m, *W3m, *b3m, *d1m, *d2m, *d4m;
  const float *W1p, *b1p, *W2p, *b2p, *W3p, *b3p, *d1p, *d2p, *d4p;
  float* out;
  int B;
};

__device__ __forceinline__ float warp_sum32(float v) {
  v += __shfl_down(v, 16, 32);
  v += __shfl_down(v, 8, 32);
  v += __shfl_down(v, 4, 32);
  v += __shfl_down(v, 2, 32);
  v += __shfl_down(v, 1, 32);
  return v;
}

// GEMV tile via V_WMMA_F32_16X16X4_F32.
// A row 0 = x[k0..k0+3] (rows 1..15 zeroed arithmetically via `sel`, so the
// LDS reads stay unconditional broadcasts -- no EXEC masking in the hot loop).
// B[k][n] = W[tile*16+n][k0+k].  D[0][n] = partial y[tile*16+n], lane n, acc[0].
__device__ __forceinline__ float gemv_tile_wmma(const float* __restrict__ W, int ld,
                                                int kdim, const float* __restrict__ xin,
                                                int tile, int lane) {
  const int n  = lane & 15;
  const int hi = lane >> 4;                 // 0: K=0,1  1: K=2,3
  const float sel = (n == 0) ? 1.0f : 0.0f; // arithmetic row-0 mask
  const float* wrow = W + (size_t)(tile * 16 + n) * (size_t)ld + (hi << 1);
  const float* xrow = xin + (hi << 1);
  v8f acc0 = {0.f, 0.f, 0.f, 0.f, 0.f, 0.f, 0.f, 0.f};
  v8f acc1 = {0.f, 0.f, 0.f, 0.f, 0.f, 0.f, 0.f, 0.f};
  const int nk = kdim >> 2;                 // 98 or 128 (always even)
  for (int kc = 0; kc < nk; kc += 2) {
    {
      float2 xv = *reinterpret_cast<const float2*>(xrow + (kc << 2));
      v2f a; a.x = xv.x * sel; a.y = xv.y * sel;
      float2 wb = *reinterpret_cast<const float2*>(wrow + (kc << 2));
      v2f bb; bb.x = wb.x; bb.y = wb.y;
      acc0 = __builtin_amdgcn_wmma_f32_16x16x4_f32(false, a, false, bb,
                                                   (short)0, acc0, false, false);
    }
    {
      float2 xv = *reinterpret_cast<const float2*>(xrow + ((kc + 1) << 2));
      v2f a; a.x = xv.x * sel; a.y = xv.y * sel;
      float2 wb = *reinterpret_cast<const float2*>(wrow + ((kc + 1) << 2));
      v2f bb; bb.x = wb.x; bb.y = wb.y;
      acc1 = __builtin_amdgcn_wmma_f32_16x16x4_f32(false, a, false, bb,
                                                   (short)0, acc1, false, false);
    }
  }
  return acc0[0] + acc1[0];                 // valid on lanes 0..15
}

template <int OHK>
__device__ void stage_core(
    int t, int w, int lane,
    const float* __restrict__ pe, const float* __restrict__ po,
    const float* __restrict__ ye, const float* __restrict__ yo,
    float d1, float d2, float d4,
    const float* __restrict__ W1l, const float* __restrict__ b1l,
    const float* __restrict__ W2l, const float* __restrict__ b2l,
    const float* __restrict__ W3l, const float* __restrict__ b3l,
    const float* __restrict__ mapp,
    float* s_x, float* s_c, float* s_s,
    float* s_P, float* s_coef, float* s_r, float* s_in, float* s_z,
    float* s_oh, float* s_v, float* xp, float* __restrict__ gout) {
  // (a) P = [Psi_e ; Psi_o] @ xvec   (1024 rows, coalesced float4 streaming)
  {
    const float4* x4 = reinterpret_cast<const float4*>(s_x);
    for (int vr = w; vr < 2 * D_PR; vr += 8) {
      const float* rowp = (vr < D_PR) ? (pe + (size_t)vr * D_2SL)
                                      : (po + (size_t)(vr - D_PR) * D_2SL);
      const float4* r4 = reinterpret_cast<const float4*>(rowp);
      float acc = 0.f;
      for (int kk = lane; kk < D_2SL / 4; kk += 32) {
        float4 p = r4[kk], xx = x4[kk];
        acc = fmaf(p.x, xx.x, acc); acc = fmaf(p.y, xx.y, acc);
        acc = fmaf(p.z, xx.z, acc); acc = fmaf(p.w, xx.w, acc);
      }
      acc = warp_sum32(acc);
      if (lane == 0) s_P[vr] = acc;
    }
  }
  __syncthreads();
  // (b) u and packed g/h coefficients
  {
    const int k = t;
    float pe0 = s_P[k], pe1 = s_P[k + D_BL];
    float po0 = s_P[2 * D_BL + k], po1 = s_P[3 * D_BL + k];
    float we = pe0 * pe0 + pe1 * pe1;
    float wo = po0 * po0 + po1 * po1;
    float ue = fmaf(d2, we, -d1 * ye[k]);
    float uo = fmaf(d4, wo, -d2 * yo[k]);
    s_coef[k] = pe0 * ue;            s_coef[D_BL + k] = pe1 * ue;
    s_coef[2 * D_BL + k] = po0 * uo; s_coef[3 * D_BL + k] = po1 * uo;
  }
  __syncthreads();
  // (c) r[j] = sum_m Psi_e[m][j]*coef_e[m] + Psi_o[m][j]*coef_o[m] (coalesced float2)
  {
    const float2* pe2 = reinterpret_cast<const float2*>(pe);
    const float2* po2 = reinterpret_cast<const float2*>(po);
    for (int jp = t; jp < D_SL; jp += NTHR) {   // 264 column pairs
      float a0 = 0.f, a1 = 0.f;
      for (int m = 0; m < D_PR; ++m) {
        float2 u = pe2[m * (D_2SL / 2) + jp];
        float cc = s_coef[m];
        a0 = fmaf(u.x, cc, a0); a1 = fmaf(u.y, cc, a1);
      }
      for (int m = 0; m < D_PR; ++m) {
        float2 u = po2[m * (D_2SL / 2) + jp];
        float cc = s_coef[2 * D_BL + m];
        a0 = fmaf(u.x, cc, a0); a1 = fmaf(u.y, cc, a1);
      }
      s_r[2 * jp] = a0; s_r[2 * jp + 1] = a1;
    }
  }
  __syncthreads();
  // (d) q = x_prev + c*r[:SL] + s*r[SL:];  s_in = cat(q, v)
  for (int j = t; j < D_SL; j += NTHR)
    s_in[j] = xp[j] + s_c[j] * s_r[j] + s_s[j] * s_r[j + D_SL];
  for (int i = t; i < D_VL; i += NTHR) s_in[D_SL + i] = s_v[i];
  __syncthreads();
  // (e) z = hardtanh(W1 @ s_in + b1)   [WMMA]
  for (int tt = 0; tt < 4; ++tt) {           // 32 tiles of 16 over 8 waves
    int tile = w + 8 * tt;
    float y = gemv_tile_wmma(W1l, D_IN1, D_IN1, s_in, tile, lane);
    if (lane < 16) {
      int o = tile * 16 + lane;
      float zz = y + b1l[o];
      s_z[o] = fminf(fmaxf(zz, -10.f), 10.f);
    }
  }
  __syncthreads();
  // (f) oh += W2 @ z + b2 ;  v += W3 @ z + b3   [WMMA]
  {
    const int OHTILES = D_SL * OHK / 16;     // 66 or 132
    for (int tile = w; tile < OHTILES; tile += 8) {
      float y = gemv_tile_wmma(W2l, D_ZL, D_ZL, s_z, tile, lane);
      if (lane < 16) { int o = tile * 16 + lane; s_oh[o] += y + b2l[o]; }
    }
    {
      int tile = w;                          // 8 tiles, one per wave
      float y = gemv_tile_wmma(W3l, D_ZL, D_ZL, s_z, tile, lane);
      if (lane < 16) { int o = tile * 16 + lane; s_v[o] += y + b3l[o]; }
    }
  }
  __syncthreads();
  // (g) oh2sym: softmax over OHK bins, weighted by mapp; write new x and output
  for (int s2 = t; s2 < D_SL; s2 += NTHR) {
    const float* row = s_oh + s2 * OHK;
    float mx = row[0];
#pragma unroll
    for (int k = 1; k < OHK; ++k) mx = fmaxf(mx, row[k]);
    float den = 0.f, num = 0.f;
#pragma unroll
    for (int k = 0; k < OHK; ++k) {
      float e = expf(row[k] - mx);
      den += e; num = fmaf(e, mapp[k], num);
    }
    float val = num / den;
    xp[s2] = val;
    gout[s2] = val;
  }
}

__global__ void __launch_bounds__(NTHR) detnet_kernel(Args a) {
  __shared__ __align__(16) float s_x[D_2SL];
  __shared__ __align__(16) float s_r[D_2SL];
  __shared__ __align__(16) float s_in[D_IN1];
  __shared__ __align__(16) float s_z[D_ZL];
  __shared__ float s_c[D_SL], s_s[D_SL], s_cos[D_SL], s_sin[D_SL];
  __shared__ float s_mag[D_SL], s_ph[D_SL];
  __shared__ float s_P[2 * D_PR];
  __shared__ float s_coef[2 * D_PR];
  __shared__ float s_oh_m[D_SL * 4];
  __shared__ float s_oh_p[D_SL * 8];
  __shared__ float s_v_m[D_VL], s_v_p[D_VL];

  const int t = threadIdx.x;
  const int w = t >> 5, lane = t & 31;
  const int b = blockIdx.x;
  const int B = a.B;

  const float* pe = a.Psi_e + (size_t)b * D_PR * D_2SL;
  const float* po = a.Psi_o + (size_t)b * D_PR * D_2SL;
  const float* ye = a.y_e + (size_t)b * D_BL;
  const float* yo = a.y_o + (size_t)b * D_BL;
  float* outm = a.out;
  float* outp = a.out + (size_t)D_LAYERS * B * D_SL;

  for (int i = t; i < D_SL; i += NTHR) { s_mag[i] = 0.f; s_ph[i] = 0.f; }
  for (int i = t; i < D_SL * 4; i += NTHR) s_oh_m[i] = 0.f;
  for (int i = t; i < D_SL * 8; i += NTHR) s_oh_p[i] = 0.f;
  for (int i = t; i < D_VL; i += NTHR) { s_v_m[i] = 0.f; s_v_p[i] = 0.f; }
  __syncthreads();

#pragma unroll 1
  for (int l = 0; l < D_LAYERS; ++l) {
    for (int j = t; j < D_SL; j += NTHR) {
      float p = s_ph[j];
      s_cos[j] = cosf(p); s_sin[j] = sinf(p);
    }
    __syncthreads();
    // stage 0 (magnitude): xvec = [mag*cos, mag*sin], c=cos, s=sin
    for (int j = t; j < D_SL; j += NTHR) {
      float m = s_mag[j];
      s_x[j] = m * s_cos[j];
      s_x[D_SL + j] = m * s_sin[j];
      s_c[j] = s_cos[j];
      s_s[j] = s_sin[j];
    }
    __syncthreads();
    stage_core<4>(t, w, lane, pe, po, ye, yo,
                  a.d1m[l], a.d2m[l], a.d4m[l],
                  a.W1m + (size_t)l * D_ZL * D_IN1, a.b1m + (size_t)l * D_ZL,
                  a.W2m + (size_t)l * D_SL * 4 * D_ZL, a.b2m + (size_t)l * D_SL * 4,
                  a.W3m + (size_t)l * D_VL * D_ZL, a.b3m + (size_t)l * D_VL,
                  a.mapp_mag,
                  s_x, s_c, s_s, s_P, s_coef, s_r, s_in, s_z,
                  s_oh_m, s_v_m, s_mag,
                  outm + ((size_t)l * B + b) * D_SL);
    __syncthreads();
    // stage 1 (phase): xvec2 = [mag_n*cos, mag_n*sin], c=-sin*mag_n, s=cos*mag_n
    for (int j = t; j < D_SL; j += NTHR) {
      float m = s_mag[j];   // updated magnitude
      s_x[j] = m * s_cos[j];
      s_x[D_SL + j] = m * s_sin[j];
      s_c[j] = -s_sin[j] * m;
      s_s[j] = s_cos[j] * m;
    }
    __syncthreads();
    stage_core<8>(t, w, lane, pe, po, ye, yo,
                  a.d1p[l], a.d2p[l], a.d4p[l],
                  a.W1p + (size_t)l * D_ZL * D_IN1, a.b1p + (size_t)l * D_ZL,
                  a.W2p + (size_t)l * D_SL * 8 * D_ZL, a.b2p + (size_t)l * D_SL * 8,
                  a.W3p + (size_t)l * D_VL * D_ZL, a.b3p + (size_t)l * D_VL,
                  a.mapp_phase,
                  s_x, s_c, s_s, s_P, s_coef, s_r, s_in, s_z,
                  s_oh_p, s_v_p, s_ph,
                  outp + ((size_t)l * B + b) * D_SL);
    __syncthreads();
  }
}

extern "C" void kernel_launch(void* const* d_in, const int* in_sizes, int n_in,
                              void* d_out, int out_size, void* d_ws, size_t ws_size,
                              hipStream_t stream) {
  (void)d_ws; (void)ws_size; (void)out_size;
  // Classify inputs by element count (robust to insertion-order or sorted flattening).
  const float* Psi[2] = {nullptr, nullptr}; int nPsi = 0;
  const float* Y[2]   = {nullptr, nullptr}; int nY = 0;
  const float* W1s[2] = {nullptr, nullptr}; int nW1 = 0;
  const float* b1s[2] = {nullptr, nullptr}; int nb1 = 0;
  const float* W3s[2] = {nullptr, nullptr}; int nW3 = 0;
  const float* b3s[2] = {nullptr, nullptr}; int nb3 = 0;
  const float* dPtr[6] = {nullptr}; int dIdx[6] = {0}; int nd = 0;
  const float *mapp4 = nullptr, *mapp8 = nullptr;
  const float *W2m = nullptr, *W2p = nullptr, *b2m = nullptr, *b2p = nullptr;
  int ySize = D_BL * D_BL;
  for (int i = 0; i < n_in; ++i) {
    const int s = in_sizes[i];
    const float* p = (const float*)d_in[i];
    if (s == 69206016)      { if (nPsi < 2) Psi[nPsi++] = p; }
    else if (s == 65536)    { if (nY < 2) { Y[nY++] = p; ySize = s; } }
    else if (s == 4)        mapp4 = p;
    else if (s == 8)        mapp8 = p;
    else if (s == 2007040)  { if (nW1 < 2) W1s[nW1++] = p; }
    else if (s == 5120)     { if (nb1 < 2) b1s[nb1++] = p; }
    else if (s == 5406720)  W2m = p;
    else if (s == 10813440) W2p = p;
    else if (s == 10560)    b2m = p;
    else if (s == 21120)    b2p = p;
    else if (s == 655360)   { if (nW3 < 2) W3s[nW3++] = p; }
    else if (s == 1280)     { if (nb3 < 2) b3s[nb3++] = p; }
    else if (s == 10)       { if (nd < 6) { dPtr[nd] = p; dIdx[nd] = i; nd++; } }
  }
  Args a;
  a.y_e = Y[0]; a.y_o = Y[1];
  a.Psi_e = Psi[0]; a.Psi_o = Psi[1];
  a.mapp_mag = mapp4; a.mapp_phase = mapp8;
  a.W1m = W1s[0]; a.W1p = W1s[1];
  a.b1m = b1s[0]; a.b1p = b1s[1];
  a.W2m = W2m; a.W2p = W2p; a.b2m = b2m; a.b2p = b2p;
  a.W3m = W3s[0]; a.W3p = W3s[1];
  a.b3m = b3s[0]; a.b3p = b3s[1];
  // Six size-10 scalars: contiguous run of 6 => sorted order (d1m,d1p,d2m,d2p,d4m,d4p);
  // otherwise two triples (d1m,d2m,d4m then d1p,d2p,d4p).
  if (nd == 6 && dIdx[3] == dIdx[0] + 3) {
    a.d1m = dPtr[0]; a.d1p = dPtr[1];
    a.d2m = dPtr[2]; a.d2p = dPtr[3];
    a.d4m = dPtr[4]; a.d4p = dPtr[5];
  } else {
    a.d1m = dPtr[0]; a.d2m = dPtr[1]; a.d4m = dPtr[2];
    a.d1p = dPtr[3]; a.d2p = dPtr[4]; a.d4p = dPtr[5];
  }
  a.out = (float*)d_out;
  const int B = ySize / D_BL;   // 256
  a.B = B;
  detnet_kernel<<<dim3(B), dim3(NTHR), 0, stream>>>(a);
}